// GCN_4_44805098832496
// MI455X (gfx1250) — compile-verified
//
#include <hip/hip_runtime.h>

typedef __attribute__((ext_vector_type(2))) float v2f;
typedef __attribute__((ext_vector_type(8))) float v8f;

#define N_NODES   100000
#define HIDDEN    20
#define N_FEAT    512
#define N_CLASSES 10
#define N_EDGES   3200000

// ---------------------------------------------------------------------------
// Degree: deg[col[e]] += 1
// ---------------------------------------------------------------------------
__global__ void deg_kernel(const int* __restrict__ col, float* __restrict__ deg) {
  int t = blockIdx.x * blockDim.x + threadIdx.x;
  if (t < N_EDGES) atomicAdd(&deg[col[t]], 1.0f);
}

// ---------------------------------------------------------------------------
// norm[e] = dis[row[e]] * dis[col[e]],  dis = deg>0 ? rsqrt(max(deg,1)) : 0
// ---------------------------------------------------------------------------
__global__ void norm_kernel(const int* __restrict__ row, const int* __restrict__ col,
                            const float* __restrict__ deg, float* __restrict__ norm) {
  int t = blockIdx.x * blockDim.x + threadIdx.x;
  if (t >= N_EDGES) return;
  float dr = deg[row[t]];
  float dc = deg[col[t]];
  float ir = dr > 0.f ? rsqrtf(fmaxf(dr, 1.f)) : 0.f;
  float ic = dc > 0.f ? rsqrtf(fmaxf(dc, 1.f)) : 0.f;
  norm[t] = ir * ic;
}

// ---------------------------------------------------------------------------
// C[nRows x 20] = A[nRows x K] @ W[K x 20]   (no bias, no relu)
// WMMA f32 16x16x4. Block = 256 threads (8 waves); each wave owns a
// 16-row x 32-col tile (2 accumulators).
// LDS layout for W is k-pair interleaved: element (k,c) lives at
// ((k>>1)*32 + c)*2 + (k&1), so each B fragment {B[k][c], B[k+1][c]} is a
// single contiguous ds_load_b64 into an even-aligned VGPR pair.
// ---------------------------------------------------------------------------
template <int K>
__global__ void gemm20_wmma(const float* __restrict__ A, const float* __restrict__ W,
                            float* __restrict__ C, int nRows) {
  __shared__ float ldsW[K * 32];  // cols 20..31 zero-padded

  int tid = threadIdx.x;
  for (int i = tid; i < K * 32; i += 256) {
    int r = i & 1;
    int c = (i >> 1) & 31;
    int p = i >> 6;
    int k = (p << 1) | r;
    ldsW[i] = (c < HIDDEN) ? W[k * HIDDEN + c] : 0.f;
  }
  __syncthreads();

  int wave  = tid >> 5;
  int lane  = tid & 31;
  int laneM = lane & 15;          // row (A) / col (B) within tile
  int kOff  = (lane >> 4) << 1;   // lanes 0-15: K=+0,+1  lanes 16-31: K=+2,+3

  int rowBase = blockIdx.x * 128 + wave * 16;
  int arow = rowBase + laneM;
  if (arow >= nRows) arow = nRows - 1;        // clamp (stores are guarded)
  const float* Ar = A + (size_t)arow * K;     // row base 16B aligned for K=512/20

  v8f acc0 = {};
  v8f acc1 = {};
#pragma unroll 4
  for (int k = 0; k < K; k += 4) {
    int p = (k + kOff) >> 1;                  // k+kOff is even
    v2f a  = *(const v2f*)(Ar + k + kOff);                    // global_load_b64
    v2f b0 = *(const v2f*)(ldsW + (p * 32 + laneM) * 2);      // ds_load_b64
    v2f b1 = *(const v2f*)(ldsW + (p * 32 + 16 + laneM) * 2); // ds_load_b64
    acc0 = __builtin_amdgcn_wmma_f32_16x16x4_f32(false, a, false, b0,
                                                 (short)0, acc0, false, false);
    acc1 = __builtin_amdgcn_wmma_f32_16x16x4_f32(false, a, false, b1,
                                                 (short)0, acc1, false, false);
  }

  // C/D layout: VGPR v, lanes 0-15 -> M=v, N=lane; lanes 16-31 -> M=v+8, N=lane-16
  int mOff = (lane >> 4) << 3;
#pragma unroll
  for (int v = 0; v < 8; ++v) {
    int r = rowBase + mOff + v;
    if (r < nRows) {
      C[(size_t)r * HIDDEN + laneM] = acc0[v];
      if (laneM < HIDDEN - 16)                    // cols 16..19 only
        C[(size_t)r * HIDDEN + 16 + laneM] = acc1[v];
    }
  }
}

// ---------------------------------------------------------------------------
// agg[col[e]] += norm[e] * hlin[row[e]]   (thread = one float4 quad of one edge)
// ---------------------------------------------------------------------------
__global__ void scatter_kernel(const int* __restrict__ row, const int* __restrict__ col,
                               const float* __restrict__ norm,
                               const float* __restrict__ hlin, float* __restrict__ agg) {
  int t = blockIdx.x * blockDim.x + threadIdx.x;
  if (t >= N_EDGES * 5) return;
  int e = t / 5;
  int q = t - e * 5;
  int r = row[e];
  int c = col[e];
  float w = norm[e];
  float4 v = reinterpret_cast<const float4*>(hlin + (size_t)r * HIDDEN)[q];
  float* dst = agg + (size_t)c * HIDDEN + q * 4;
  atomicAdd(dst + 0, w * v.x);
  atomicAdd(dst + 1, w * v.y);
  atomicAdd(dst + 2, w * v.z);
  atomicAdd(dst + 3, w * v.w);
}

// ---------------------------------------------------------------------------
// h = relu(agg + b)
// ---------------------------------------------------------------------------
__global__ void bias_relu_kernel(const float* __restrict__ agg, const float* __restrict__ b,
                                 float* __restrict__ h) {
  int t = blockIdx.x * blockDim.x + threadIdx.x;
  if (t >= N_NODES * HIDDEN) return;
  int d = t % HIDDEN;
  h[t] = fmaxf(agg[t] + b[d], 0.f);
}

// ---------------------------------------------------------------------------
// out = log_softmax(concat(h1..h4) @ Wp + bp)   (one thread per node)
// ---------------------------------------------------------------------------
__global__ void final_kernel(const float* __restrict__ h1, const float* __restrict__ h2,
                             const float* __restrict__ h3, const float* __restrict__ h4,
                             const float* __restrict__ Wp, const float* __restrict__ bp,
                             float* __restrict__ out) {
  __shared__ float sW[4 * HIDDEN * N_CLASSES];
  __shared__ float sB[N_CLASSES];
  int tid = threadIdx.x;
  for (int i = tid; i < 4 * HIDDEN * N_CLASSES; i += blockDim.x) sW[i] = Wp[i];
  if (tid < N_CLASSES) sB[tid] = bp[tid];
  __syncthreads();

  int n = blockIdx.x * blockDim.x + tid;
  if (n >= N_NODES) return;

  float feat[4 * HIDDEN];
  const float* hs[4] = {h1, h2, h3, h4};
#pragma unroll
  for (int l = 0; l < 4; ++l)
    for (int d = 0; d < HIDDEN; ++d)
      feat[l * HIDDEN + d] = hs[l][(size_t)n * HIDDEN + d];

  float logit[N_CLASSES];
#pragma unroll
  for (int c = 0; c < N_CLASSES; ++c) {
    float s = sB[c];
    for (int i = 0; i < 4 * HIDDEN; ++i) s += feat[i] * sW[i * N_CLASSES + c];
    logit[c] = s;
  }
  float m = logit[0];
#pragma unroll
  for (int c = 1; c < N_CLASSES; ++c) m = fmaxf(m, logit[c]);
  float sum = 0.f;
#pragma unroll
  for (int c = 0; c < N_CLASSES; ++c) sum += __expf(logit[c] - m);
  float lse = m + __logf(sum);
#pragma unroll
  for (int c = 0; c < N_CLASSES; ++c) out[(size_t)n * N_CLASSES + c] = logit[c] - lse;
}

// ---------------------------------------------------------------------------
extern "C" void kernel_launch(void* const* d_in, const int* in_sizes, int n_in,
                              void* d_out, int out_size, void* d_ws, size_t ws_size,
                              hipStream_t stream) {
  (void)in_sizes; (void)n_in; (void)out_size; (void)ws_size;

  const float* x  = (const float*)d_in[0];
  const int*   ei = (const int*)d_in[1];          // [2, E]
  const int*   row = ei;
  const int*   col = ei + N_EDGES;
  const float* Ws[4] = {(const float*)d_in[2], (const float*)d_in[4],
                        (const float*)d_in[6], (const float*)d_in[8]};
  const float* bs[4] = {(const float*)d_in[3], (const float*)d_in[5],
                        (const float*)d_in[7], (const float*)d_in[9]};
  const float* Wp = (const float*)d_in[10];
  const float* bp = (const float*)d_in[11];
  float* out = (float*)d_out;

  // workspace carve-out (all offsets 16B-aligned)
  float* ws   = (float*)d_ws;
  float* deg  = ws;  ws += N_NODES;
  float* norm = ws;  ws += N_EDGES;
  float* hlin = ws;  ws += (size_t)N_NODES * HIDDEN;
  float* agg  = ws;  ws += (size_t)N_NODES * HIDDEN;
  float* h[4];
  for (int i = 0; i < 4; ++i) { h[i] = ws; ws += (size_t)N_NODES * HIDDEN; }

  const int TB = 256;
  hipMemsetAsync(deg, 0, N_NODES * sizeof(float), stream);
  deg_kernel<<<(N_EDGES + TB - 1) / TB, TB, 0, stream>>>(col, deg);
  norm_kernel<<<(N_EDGES + TB - 1) / TB, TB, 0, stream>>>(row, col, deg, norm);

  int gemmGrid = (N_NODES + 127) / 128;
  for (int l = 0; l < 4; ++l) {
    if (l == 0)
      gemm20_wmma<N_FEAT><<<gemmGrid, TB, 0, stream>>>(x, Ws[0], hlin, N_NODES);
    else
      gemm20_wmma<HIDDEN><<<gemmGrid, TB, 0, stream>>>(h[l - 1], Ws[l], hlin, N_NODES);
    hipMemsetAsync(agg, 0, (size_t)N_NODES * HIDDEN * sizeof(float), stream);
    scatter_kernel<<<(N_EDGES * 5 + TB - 1) / TB, TB, 0, stream>>>(row, col, norm, hlin, agg);
    bias_relu_kernel<<<(N_NODES * HIDDEN + TB - 1) / TB, TB, 0, stream>>>(agg, bs[l], h[l]);
  }

  final_kernel<<<(N_NODES + TB - 1) / TB, TB, 0, stream>>>(h[0], h[1], h[2], h[3], Wp, bp, out);
}